// NeighborhoodSimilarity_87832081203328
// MI455X (gfx1250) — compile-verified
//
#include <hip/hip_runtime.h>
#include <hip/hip_bf16.h>

typedef __attribute__((ext_vector_type(2))) float v2f;
typedef __attribute__((ext_vector_type(8))) float v8f;

#define FEAT_DIM 128

// ---------------------------------------------------------------------------
// Kernel 0: zero a float buffer
// ---------------------------------------------------------------------------
__global__ void zero_kernel(float* __restrict__ p, long long n) {
    long long i = (long long)blockIdx.x * blockDim.x + threadIdx.x;
    if (i < n) p[i] = 0.0f;
}

// ---------------------------------------------------------------------------
// Kernel 1: per-node inverse clamped norm.  One wave (32 lanes) per node:
// each lane loads one float4 (32*4 = 128 floats), squares, wave-reduces.
// ---------------------------------------------------------------------------
__global__ void invnorm_kernel(const float* __restrict__ feat,
                               float* __restrict__ invnorm, int N) {
    const int lane = threadIdx.x & 31;
    const int wavesPerBlock = blockDim.x >> 5;
    const int node = blockIdx.x * wavesPerBlock + (threadIdx.x >> 5);
    if (node >= N) return;

    const float4* row = (const float4*)(feat + (size_t)node * FEAT_DIM);
    float4 v = row[lane];
    float s = v.x * v.x + v.y * v.y + v.z * v.z + v.w * v.w;
#pragma unroll
    for (int off = 16; off > 0; off >>= 1) s += __shfl_xor(s, off, 32);

    if (lane == 0) {
        float nrm = fmaxf(sqrtf(s), 1e-8f);
        invnorm[node] = 1.0f / nrm;
    }
}

// ---------------------------------------------------------------------------
// Kernel 2: edge similarity via f32 WMMA (exact f32 FMA, matches reference).
//
// Each wave handles a tile of 16 edges.  We compute D = A * B where
//   A = 16 x 128 : rows are feat[src[e]]           (M = edge-in-tile)
//   B = 128 x 16 : cols are feat[dst[e]]           (N = edge-in-tile)
// using 32 chained V_WMMA_F32_16X16X4_F32 (K = 4 each).  Only diag(D) is
// used: D[i][i] = dot(feat[src_i], feat[dst_i]).
//
// A 16x4 layout (wave32): VGPR0 lanes0-15 K=0 / lanes16-31 K=2; VGPR1 K=1/K=3
//   -> lane (M = lane&15) loads floats [kbase + 2*(lane>>4) .. +1]
// B 4x16 layout is the symmetric transpose (lane = N), identical addressing.
//
// D layout: VGPR j: lanes0-15 -> (M=j, N=lane); lanes16-31 -> (M=j+8, N=lane-16)
//   diag i<8  -> c[i]   at lane i
//   diag i>=8 -> c[i-8] at lane i+16  (i.e. lanes 24..31)
// Those lanes' (sub = lane&15) src/dst are exactly that edge's endpoints.
// ---------------------------------------------------------------------------
__global__ void edge_sim_kernel(const float* __restrict__ feat,
                                const long long* __restrict__ ei, // [2, E]
                                const float* __restrict__ invnorm,
                                float* __restrict__ sums,
                                float* __restrict__ deg,
                                long long E) {
    const int lane = threadIdx.x & 31;
    const int wavesPerBlock = blockDim.x >> 5;
    const long long tileBase =
        ((long long)blockIdx.x * wavesPerBlock + (threadIdx.x >> 5)) * 16;
    if (tileBase >= E) return;  // wave-uniform

    const int sub  = lane & 15;  // edge-in-tile: A row (M) / B col (N)
    const int half = lane >> 4;  // K sub-offset selector

    long long e = tileBase + sub;
    const bool valid = (e < E);
    if (!valid) e = E - 1;  // clamp for safe loads; masked at atomic stage

    const int s = (int)ei[e];
    const int d = (int)ei[E + e];
    const float* __restrict__ srow = feat + (size_t)s * FEAT_DIM;
    const float* __restrict__ drow = feat + (size_t)d * FEAT_DIM;

    v8f c = {0.f, 0.f, 0.f, 0.f, 0.f, 0.f, 0.f, 0.f};
#pragma unroll
    for (int t = 0; t < FEAT_DIM / 4; ++t) {
        const int k = t * 4 + half * 2;         // even -> 8B aligned
        v2f a = *(const v2f*)(srow + k);        // A: (M=sub, K=k, k+1)
        v2f b = *(const v2f*)(drow + k);        // B: (K=k, k+1, N=sub)
        c = __builtin_amdgcn_wmma_f32_16x16x4_f32(
            /*neg_a=*/false, a, /*neg_b=*/false, b,
            /*c_mod=*/(short)0, c, /*reuse_a=*/false, /*reuse_b=*/false);
    }

    // Extract diagonal: lane j (j<8) gets edge j, lane j+24 gets edge j+8.
    float sim = 0.0f;
#pragma unroll
    for (int j = 0; j < 8; ++j) {
        float v = c[j];
        if (lane == j)      sim = v;
        if (lane == j + 24) sim = v;
    }

    const bool owner = (lane < 8) || (lane >= 24);
    if (owner && valid) {
        const float val = sim * invnorm[s] * invnorm[d];
        atomicAdd(&sums[s], val);
        atomicAdd(&sums[d], val);
        atomicAdd(&deg[s], 1.0f);
        atomicAdd(&deg[d], 1.0f);
    }
}

// ---------------------------------------------------------------------------
// Kernel 3: finalize   out[i] = deg>0 ? sum/deg : 1.0
// ---------------------------------------------------------------------------
__global__ void finalize_kernel(const float* __restrict__ sums,
                                const float* __restrict__ deg,
                                float* __restrict__ out, int N) {
    int i = blockIdx.x * blockDim.x + threadIdx.x;
    if (i < N) {
        float dg = deg[i];
        out[i] = (dg > 0.0f) ? (sums[i] / dg) : 1.0f;
    }
}

// ---------------------------------------------------------------------------
// Launcher
// ---------------------------------------------------------------------------
extern "C" void kernel_launch(void* const* d_in, const int* in_sizes, int n_in,
                              void* d_out, int out_size, void* d_ws, size_t ws_size,
                              hipStream_t stream) {
    const float*     feat = (const float*)d_in[0];       // [N, 128] f32
    const long long* ei   = (const long long*)d_in[1];   // [2, E]  i64

    const int       N = out_size;                 // 100000
    const long long E = (long long)in_sizes[1] / 2;

    // Workspace layout: invnorm[N] | sums[N] | deg[N]
    float* invnorm = (float*)d_ws;
    float* sums    = invnorm + N;
    float* deg     = sums + N;

    // Zero accumulators (sums + deg are contiguous)
    {
        long long n = 2LL * N;
        long long blocks = (n + 255) / 256;
        zero_kernel<<<(dim3)(unsigned)blocks, 256, 0, stream>>>(sums, n);
    }

    // Inverse norms: one wave per node, 8 waves per 256-thread block
    {
        int blocks = (N + 7) / 8;
        invnorm_kernel<<<blocks, 256, 0, stream>>>(feat, invnorm, N);
    }

    // Edge similarities: 16 edges per wave, 8 waves (128 edges) per block
    {
        long long tiles  = (E + 15) / 16;
        long long blocks = (tiles + 7) / 8;
        edge_sim_kernel<<<(dim3)(unsigned)blocks, 256, 0, stream>>>(
            feat, ei, invnorm, sums, deg, E);
    }

    // Finalize
    {
        int blocks = (N + 255) / 256;
        finalize_kernel<<<blocks, 256, 0, stream>>>(sums, deg, (float*)d_out, N);
    }
}